// TopPool_38920993636619
// MI455X (gfx1250) — compile-verified
//
#include <hip/hip_runtime.h>
#include <hip/hip_bf16.h>

// TopPool: out[b,c,h,w] = max_{h' >= h} x[b,c,h',w]  (reverse cummax over H)
// x: (16, 256, 128, 128) fp32. Pure streaming scan: 268 MB in + 268 MB out,
// ~23 us at 23.3 TB/s HBM. One wave32 per (b,c) plane, one float4 of w per lane.

typedef __attribute__((ext_vector_type(4))) float v4f;

#ifndef INFINITY
#define INFINITY __builtin_inff()
#endif

static __device__ __forceinline__ v4f vmax4(v4f a, v4f b) {
    v4f r;
#pragma unroll
    for (int i = 0; i < 4; ++i) r[i] = fmaxf(a[i], b[i]);   // v_max_num_f32
    return r;
}

namespace {
constexpr int kH      = 128;
constexpr int kW      = 128;
constexpr int kPlanes = 16 * 256;       // (b,c) planes, one wave32 each
constexpr int kChunk  = 8;              // H rows per unrolled chunk (MLP depth)
constexpr int kRowV4  = kW / 4;         // 32 float4 per H row
}

__global__ __launch_bounds__(256) void TopPool_38920993636619_kernel(
    const float* __restrict__ x, float* __restrict__ out) {
    const int gtid = blockIdx.x * blockDim.x + threadIdx.x;
    const int wave = gtid >> 5;                 // wave32: one (b,c) plane per wave
    const int lane = threadIdx.x & 31;          // lane owns w = [4*lane, 4*lane+3]
    if (wave >= kPlanes) return;

    const size_t planeOff = (size_t)wave * (kH * kW);
    const v4f* __restrict__ src = (const v4f*)(x + planeOff);
    v4f* __restrict__       dst = (v4f*)(out + planeOff);
    const char* __restrict__ pf = (const char*)(x + planeOff);

    v4f run;
#pragma unroll
    for (int i = 0; i < 4; ++i) run[i] = -INFINITY;

    // Walk H from 127 down to 0 in chunks of 8 rows.
    // Each chunk: prefetch next chunk (1 insn covers 4 KB: lane*128B = 32
    // cachelines), issue 8 independent b128 NT loads, then fold the suffix
    // max and stream 8 b128 NT stores.
    for (int h0 = kH - kChunk; h0 >= 0; h0 -= kChunk) {
        if (h0 >= kChunk) {
            __builtin_prefetch(pf + (size_t)(h0 - kChunk) * (kW * sizeof(float))
                                  + (size_t)lane * 128, 0, 1);  // global_prefetch_b8
        }

        v4f v[kChunk];
#pragma unroll
        for (int i = 0; i < kChunk; ++i)
            v[i] = __builtin_nontemporal_load(src + (size_t)(h0 + i) * kRowV4 + lane);

#pragma unroll
        for (int i = kChunk - 1; i >= 0; --i) {
            run = vmax4(run, v[i]);
            __builtin_nontemporal_store(run, dst + (size_t)(h0 + i) * kRowV4 + lane);
        }
    }
}

extern "C" void kernel_launch(void* const* d_in, const int* in_sizes, int n_in,
                              void* d_out, int out_size, void* d_ws, size_t ws_size,
                              hipStream_t stream) {
    (void)in_sizes; (void)n_in; (void)d_ws; (void)ws_size; (void)out_size;
    const float* x = (const float*)d_in[0];
    float* out     = (float*)d_out;

    const int threadsPerBlock = 256;                       // 8 wave32 per block
    const int totalThreads    = kPlanes * 32;              // one wave per plane
    const int blocks          = totalThreads / threadsPerBlock;  // 512

    TopPool_38920993636619_kernel<<<blocks, threadsPerBlock, 0, stream>>>(x, out);
}